// GraphAggregator_29970281791938
// MI455X (gfx1250) — compile-verified
//
#include <hip/hip_runtime.h>

// CDNA5 / gfx1250: wave32, V_WMMA_F32_16X16X4_F32 for mean pooling.
typedef __attribute__((ext_vector_type(2))) float v2f;
typedef __attribute__((ext_vector_type(4))) float v4f;
typedef __attribute__((ext_vector_type(8))) float v8f;

__device__ __forceinline__ v8f wmma_acc_f32(v2f a, v2f b, v8f c) {
  // 8 args: (neg_a, A, neg_b, B, c_mod, C, reuse_a, reuse_b)
  return __builtin_amdgcn_wmma_f32_16x16x4_f32(
      false, a, false, b, (short)0, c, false, false);
}

// One wave computes one output row: dst[row] = (1/16) * sum_{j<16} src[nbr[node,j]].
// DEGREE = 16 (4 WMMA K-groups of 4 neighbors), EMBED = 64 (4 dims/lane).
//
// Per neighbor group j0: lanes 0-15 load the full 256B of row (s0) as b128
// (dims 4*col..4*col+3) and row (s1); lanes 16-31 load rows (s2),(s3).
// B operand i = {L0[i], L1[i]}: column n=col receives dim 4*col+i of all four
// neighbors across the two lane halves / two B VGPRs. With A == 1/16
// everywhere, D[m,col] = (1/16) * sum of the group's 4 neighbors at that dim,
// identical for every m -- chaining 4 WMMAs accumulates the full 16-neighbor
// mean. acc_i element 0 on lane col then holds dim 4*col+i -> one b128 store.
template <bool INDIRECT>
__global__ __launch_bounds__(256) void hop_mean_wmma(
    const float* __restrict__ src,        // [N, 64] rows to gather from
    const int*   __restrict__ neighbors,  // [NUM_EMBED, 16]
    const int*   __restrict__ nodes,      // [count] query ids (INDIRECT only)
    float*       __restrict__ dst,        // [count, 64]
    int count) {
  int wid = blockIdx.x * (blockDim.x >> 5) + (threadIdx.x >> 5);
  const int lane = threadIdx.x & 31;
  if (wid >= count) return;           // wave-uniform: EXEC all-ones inside
  wid = __builtin_amdgcn_readfirstlane(wid);  // force SGPR -> s_load indices

  int node = INDIRECT ? nodes[wid] : wid;
  node = __builtin_amdgcn_readfirstlane(node);
  const int* __restrict__ nbr = neighbors + (long)node * 16;  // SMEM loads

  const int half = lane >> 4;   // which neighbor pair this half-wave carries
  const int col  = lane & 15;   // output column (N) of the WMMA

  const v2f a = {0.0625f, 0.0625f};  // A(16x4) == 1/16 for all lanes/VGPRs
  v8f acc0 = {}, acc1 = {}, acc2 = {}, acc3 = {};

#pragma unroll
  for (int j0 = 0; j0 < 16; j0 += 4) {
    // Wave-uniform neighbor indices (scalar loads), per-lane pair select.
    const int s0 = nbr[j0 + 0];
    const int s1 = nbr[j0 + 1];
    const int s2 = nbr[j0 + 2];
    const int s3 = nbr[j0 + 3];
    const int r0 = half ? s2 : s0;
    const int r1 = half ? s3 : s1;

    // Two b128 loads per group: each covers one full 256B row per lane-half.
    const v4f L0 = *(const v4f*)(src + (long)r0 * 64 + 4 * col);
    const v4f L1 = *(const v4f*)(src + (long)r1 * 64 + 4 * col);

    v2f b0 = {L0[0], L1[0]};   // dim 4*col+0 of neighbors (r0, r1)
    v2f b1 = {L0[1], L1[1]};   // dim 4*col+1
    v2f b2 = {L0[2], L1[2]};   // dim 4*col+2
    v2f b3 = {L0[3], L1[3]};   // dim 4*col+3

    acc0 = wmma_acc_f32(a, b0, acc0);
    acc1 = wmma_acc_f32(a, b1, acc1);
    acc2 = wmma_acc_f32(a, b2, acc2);
    acc3 = wmma_acc_f32(a, b3, acc3);
  }

  // All D rows are identical; element 0 on lanes 0-15 carries column col.
  // Lanes 0-15 write the whole 256B row with one b128 store.
  if (half == 0) {
    v4f o = {acc0[0], acc1[0], acc2[0], acc3[0]};
    *(v4f*)(dst + (long)wid * 64 + 4 * col) = o;
  }
}

extern "C" void kernel_launch(void* const* d_in, const int* in_sizes, int n_in,
                              void* d_out, int out_size, void* d_ws, size_t ws_size,
                              hipStream_t stream) {
  const float* embed     = (const float*)d_in[0];  // [NUM_EMBED, 64] fp32
  const int*   neighbors = (const int*)d_in[1];    // [NUM_EMBED, 16] int32
  const int*   nodes     = (const int*)d_in[2];    // [NUM_QUERY] int32
  // d_in[3] = k_hop (static 2 per reference; device scalar not host-readable
  // under graph capture, so the 2-hop schedule is fixed).

  const int num_embed = in_sizes[1] / 16;
  const int num_query = in_sizes[2];

  float* out = (float*)d_out;
  float* h1  = (float*)d_ws;  // [num_embed, 64] intermediate (51.2 MB)

  const dim3 block(256);  // 8 waves per block

  // Hop 1: h1[v] = mean_j embed[neighbors[v, j]] for all nodes.
  const dim3 grid1((num_embed + 7) / 8);
  hop_mean_wmma<false><<<grid1, block, 0, stream>>>(
      embed, neighbors, nullptr, h1, num_embed);

  // Hop 2: out[q] = mean_j h1[neighbors[nodes[q], j]].
  const dim3 grid2((num_query + 7) / 8);
  hop_mean_wmma<true><<<grid2, block, 0, stream>>>(
      h1, neighbors, nodes, out, num_query);
}